// CGM_20779051778567
// MI455X (gfx1250) — compile-verified
//
#include <hip/hip_runtime.h>

// Channel-group winner-take-all, fp32, NCHW [32,512,56,56], group=4.
// Pure streaming kernel: ~411 MB traffic, HBM-bound (roofline ~17.6 us @ 23.3 TB/s).

typedef float f32x4 __attribute__((ext_vector_type(4)));

#define BATCH      32
#define CHAN       512
#define HW         3136   // 56*56
#define HW4        784    // HW / 4 (float4 chunks per channel plane)
#define CGROUPS    128    // CHAN / 4
#define N4_TOTAL   (BATCH * CGROUPS * HW4)   // 3,211,264 threads

__global__ __launch_bounds__(256) void wta_group4_kernel(
    const float* __restrict__ x, float* __restrict__ out)
{
    unsigned i = blockIdx.x * blockDim.x + threadIdx.x;
    if (i >= N4_TOTAL) return;

    unsigned hw4  = i % HW4;          // which float4 chunk in the HxW plane
    unsigned rest = i / HW4;
    unsigned grp  = rest % CGROUPS;   // channel group (4 consecutive channels)
    unsigned b    = rest / CGROUPS;

    // base element offset of channel (4*grp) at spatial position hw4*4
    size_t base = ((size_t)b * CHAN + (size_t)grp * 4u) * (size_t)HW
                + (size_t)hw4 * 4u;

    const f32x4* __restrict__ pin  = (const f32x4*)(x + base);
    f32x4*       __restrict__ pout = (f32x4*)(out + base);

    // 4 group members are HW (=3136 floats = HW4 float4s) apart.
    // Non-temporal: touch-once streams, don't pollute L2.
    f32x4 a0 = __builtin_nontemporal_load(pin + 0 * HW4);
    f32x4 a1 = __builtin_nontemporal_load(pin + 1 * HW4);
    f32x4 a2 = __builtin_nontemporal_load(pin + 2 * HW4);
    f32x4 a3 = __builtin_nontemporal_load(pin + 3 * HW4);

    f32x4 m;
    m.x = fmaxf(fmaxf(a0.x, a1.x), fmaxf(a2.x, a3.x));
    m.y = fmaxf(fmaxf(a0.y, a1.y), fmaxf(a2.y, a3.y));
    m.z = fmaxf(fmaxf(a0.z, a1.z), fmaxf(a2.z, a3.z));
    m.w = fmaxf(fmaxf(a0.w, a1.w), fmaxf(a2.w, a3.w));

    f32x4 o0, o1, o2, o3;
    o0.x = (a0.x == m.x) ? a0.x : 0.0f;
    o0.y = (a0.y == m.y) ? a0.y : 0.0f;
    o0.z = (a0.z == m.z) ? a0.z : 0.0f;
    o0.w = (a0.w == m.w) ? a0.w : 0.0f;

    o1.x = (a1.x == m.x) ? a1.x : 0.0f;
    o1.y = (a1.y == m.y) ? a1.y : 0.0f;
    o1.z = (a1.z == m.z) ? a1.z : 0.0f;
    o1.w = (a1.w == m.w) ? a1.w : 0.0f;

    o2.x = (a2.x == m.x) ? a2.x : 0.0f;
    o2.y = (a2.y == m.y) ? a2.y : 0.0f;
    o2.z = (a2.z == m.z) ? a2.z : 0.0f;
    o2.w = (a2.w == m.w) ? a2.w : 0.0f;

    o3.x = (a3.x == m.x) ? a3.x : 0.0f;
    o3.y = (a3.y == m.y) ? a3.y : 0.0f;
    o3.z = (a3.z == m.z) ? a3.z : 0.0f;
    o3.w = (a3.w == m.w) ? a3.w : 0.0f;

    __builtin_nontemporal_store(o0, pout + 0 * HW4);
    __builtin_nontemporal_store(o1, pout + 1 * HW4);
    __builtin_nontemporal_store(o2, pout + 2 * HW4);
    __builtin_nontemporal_store(o3, pout + 3 * HW4);
}

extern "C" void kernel_launch(void* const* d_in, const int* in_sizes, int n_in,
                              void* d_out, int out_size, void* d_ws, size_t ws_size,
                              hipStream_t stream) {
    (void)in_sizes; (void)n_in; (void)out_size; (void)d_ws; (void)ws_size;
    const float* x = (const float*)d_in[0];
    float* out = (float*)d_out;

    const int threads = 256;
    const int blocks = (N4_TOTAL + threads - 1) / threads;  // 12544, exact
    wta_group4_kernel<<<blocks, threads, 0, stream>>>(x, out);
}